// Slice_2095944041169
// MI455X (gfx1250) — compile-verified
//
#include <hip/hip_runtime.h>

typedef float v2f __attribute__((ext_vector_type(2)));
typedef float v8f __attribute__((ext_vector_type(8)));

#define Wimg 1024
#define Himg 1024
#define GW 16
#define GH 16
#define GD 8
#define NC 12

__device__ __forceinline__ float hatf(float t) {
    return fmaxf(0.0f, 1.0f - fabsf(t));
}

// One wave = one 16-pixel horizontal strip.
// D(16ch x 16px) = A(16x48 grid values) x B(48x16 trilinear hat weights)
// via 12 chained V_WMMA_F32_16X16X4_F32, all fp32.
// K-order: k = d + 8*m, d=depth node 0..7, m = xy node (yi*3 + xi), K=48.
// All A loads are unconditional (addresses clamped in-bounds); OOB nodes are
// zeroed through the B weights; channel rows c>=12 are garbage but never stored.
__global__ __launch_bounds__(256) void slice_wmma_kernel(
    const float* __restrict__ grid,   // [4,12,8,16,16]
    const float* __restrict__ guide,  // [4,1,1024,1024]
    float* __restrict__ out)          // [4,12,1024,1024]
{
    const int lane = threadIdx.x & 31;
    const int wave = threadIdx.x >> 5;
    const int p    = lane & 15;   // pixel within strip (N) == channel row (M)
    const int hi   = lane >> 4;   // lane half selects K pair
    const int n    = blockIdx.z;
    const int h    = (blockIdx.y << 3) + wave;
    const int w0   = blockIdx.x << 4;
    const int w    = w0 + p;

    const float SX = 16.0f / 1023.0f;

    float g  = guide[(n * Himg + h) * Wimg + w];
    float iz = 4.0f * g + 3.5f;                 // in [3.5, 7.5)
    float ix = (float)w * SX - 0.5f;
    float iy = (float)h * SX - 0.5f;

    float xbf = floorf((float)w0 * SX - 0.5f);  // strip-uniform 3-node x window
    int   xb  = (int)xbf;
    float fxp = ix - xbf;                       // in [0, 2)
    float ybf = floorf(iy);                     // strip-uniform 2-node y window
    int   yb  = (int)ybf;
    float fy  = iy - ybf;

    // ---- per-lane weight tables (compile-time indexed after unroll) ----
    float wz[GD];
#pragma unroll
    for (int d = 0; d < GD; ++d) wz[d] = hatf(iz - (float)d);

    float hx[3]; int xo[3];
#pragma unroll
    for (int xi = 0; xi < 3; ++xi) {
        int xn = xb + xi;
        bool okx = (unsigned)xn < (unsigned)GW;
        hx[xi] = okx ? hatf(fxp - (float)xi) : 0.0f;  // OOB mask folded into weight
        xo[xi] = okx ? xn : 0;                        // clamped address component
    }
    float my[2]; int yo[2];
#pragma unroll
    for (int yi = 0; yi < 2; ++yi) {
        int yn = yb + yi;
        bool oky = (unsigned)yn < (unsigned)GH;
        my[yi] = oky ? (yi ? fy : 1.0f - fy) : 0.0f;
        yo[yi] = (oky ? yn : 0) * GW;
    }
    float wxy[6];
#pragma unroll
    for (int m = 0; m < 6; ++m) {
        int yi = (m >= 3) ? 1 : 0;
        int xi = m - 3 * yi;
        wxy[m] = my[yi] * hx[xi];
    }

    // channel clamped for address safety only; rows c>=12 are never stored
    const int cc = (p < NC) ? p : (NC - 1);
    const float* gc = grid + n * (NC * GD * GH * GW) + cc * (GD * GH * GW);

    // ---- issue all 24 A loads up front (unconditional, batched) ----
    float av[12][2];
#pragma unroll
    for (int s = 0; s < 12; ++s) {
#pragma unroll
        for (int j = 0; j < 2; ++j) {
            const int k0 = 4 * s + j, k1 = k0 + 2;         // hi=0 / hi=1 K indices
            const int d0 = k0 & 7, m0 = k0 >> 3;
            const int d1 = k1 & 7, m1 = k1 >> 3;
            const int yi0 = (m0 >= 3) ? 1 : 0, xi0 = m0 - 3 * yi0;
            const int yi1 = (m1 >= 3) ? 1 : 0, xi1 = m1 - 3 * yi1;
            int off0 = d0 * (GH * GW) + yo[yi0] + xo[xi0];
            int off1 = d1 * (GH * GW) + yo[yi1] + xo[xi1];
            av[s][j] = gc[hi ? off1 : off0];
        }
    }

    // ---- 12 chained f32 WMMAs ----
    v8f acc = {};
#pragma unroll
    for (int s = 0; s < 12; ++s) {
        v2f A, B;
#pragma unroll
        for (int j = 0; j < 2; ++j) {
            const int k0 = 4 * s + j, k1 = k0 + 2;
            const int d0 = k0 & 7, m0 = k0 >> 3;
            const int d1 = k1 & 7, m1 = k1 >> 3;
            float b0 = wz[d0] * wxy[m0];
            float b1 = wz[d1] * wxy[m1];
            A[j] = av[s][j];
            B[j] = hi ? b1 : b0;
        }
        acc = __builtin_amdgcn_wmma_f32_16x16x4_f32(false, A, false, B, (short)0, acc,
                                                    false, false);
    }

    // D layout: lane holds out[c = v + 8*hi][pixel p]; only c < 12 is real
#pragma unroll
    for (int v = 0; v < 8; ++v) {
        int c = v + 8 * hi;
        if (c < NC) {
            out[((n * NC + c) * Himg + h) * Wimg + w] = acc[v];
        }
    }
}

extern "C" void kernel_launch(void* const* d_in, const int* in_sizes, int n_in,
                              void* d_out, int out_size, void* d_ws, size_t ws_size,
                              hipStream_t stream) {
    const float* grid  = (const float*)d_in[0];  // bilateral_grid [4,12,8,16,16]
    const float* guide = (const float*)d_in[1];  // guidemap [4,1,1024,1024]
    float* out = (float*)d_out;                  // [4,12,1024,1024]
    (void)in_sizes; (void)n_in; (void)out_size; (void)d_ws; (void)ws_size;

    dim3 block(256, 1, 1);                       // 8 waves = 8 rows of one 16-wide tile
    dim3 g(Wimg / 16, Himg / 8, 4);              // (64, 128, 4)
    slice_wmma_kernel<<<g, block, 0, stream>>>(grid, guide, out);
}